// PdhgTgvTorch_49503793053989
// MI455X (gfx1250) — compile-verified
//
#include <hip/hip_runtime.h>
#include <math.h>
#include <stdint.h>

// ---------------------------------------------------------------------------
// PDHG solver for TGV denoising, 768x768 f32, persistent-kernel formulation.
// SIGMA = TAU = 1 (compile-time constants in the reference).
//
// Each persistent block owns a 3-row strip. Phase 1 stages u_bar/p0/p1
// (strip + 1 halo row) into LDS using the CDNA5 Tensor Data Mover
// (TENSOR_LOAD_TO_LDS descriptors, TENSORcnt) when the builtin exists,
// falling back to per-lane GLOBAL_LOAD_ASYNC_TO_LDS_B128 (ASYNCcnt).
// Phase 2 broadcasts the two global max-norm words into LDS with
// GLOBAL_LOAD_ASYNC_TO_LDS_B32. Global max-norms use wave32 shuffle + LDS +
// one global_atomic_max_u32 per block; iterations are separated by a software
// grid barrier (agent-scope atomics, acquire/release fences, s_sleep).
//
// Workspace layout (floats), N2 = 768*768:
//   [ 0..1)N2  U      current u
//   [ 1..2)N2  UBAR   over-relaxed u
//   [ 2..4)N2  P0,P1  p components
//   [ 4..6)N2  PB0,PB1 p_bar components
//   [ 6..8)N2  VC0,VC1 UNSCALED v candidate (scale folded at read time)
//   [ 8..12)N2 W00 W01 W10 W11 UNSCALED w candidate (k = p comp, l = deriv dir)
//   then 8 uints: acc[0..1]=max|v|^2 ping-pong, acc[2..3]=max|w|^2 ping-pong,
//                 acc[4]=barrier count, acc[5]=barrier generation.
// ---------------------------------------------------------------------------

#define TGV_N    768
#define TGV_GJ   (TGV_N / 4)        // 192 float4 groups per row
#define TGV_N2   (TGV_N * TGV_N)
#define NBLOCKS  256                // 768 rows / 3 rows per block
#define NTHREADS 256
#define RPB      3                  // rows per block
#define GPB      (RPB * TGV_GJ)     // 576 float4 groups per block
#define STAGE_ROWS (RPB + 1)        // staged rows per field (strip + down halo)
#define STAGE_F  (STAGE_ROWS * TGV_N)   // 3072 floats per staged field
#define STAGE_TOT (3 * STAGE_F)         // 9216 floats = 36 KB
#define STAGE_CHUNKS (STAGE_TOT / 4)    // 2304 b128 chunks (= 9 per thread)

#if __has_builtin(__builtin_amdgcn_tensor_load_to_lds)
#define TGV_USE_TDM 1
typedef unsigned int u32x4 __attribute__((ext_vector_type(4)));
typedef int          i32x8 __attribute__((ext_vector_type(8)));
typedef int          i32x4 __attribute__((ext_vector_type(4)));
#else
#define TGV_USE_TDM 0
#endif

__device__ __forceinline__ float4 ld4(const float* b, int i, int jg) {
  return *(reinterpret_cast<const float4*>(b + (size_t)i * TGV_N) + jg);
}
__device__ __forceinline__ void st4(float* b, int i, int jg, float4 v) {
  *(reinterpret_cast<float4*>(b + (size_t)i * TGV_N) + jg) = v;
}
__device__ __forceinline__ float4 lds4(const float* p) {
  return *reinterpret_cast<const float4*>(p);
}
__device__ __forceinline__ float4 f4sub(float4 a, float4 b) {
  return make_float4(a.x - b.x, a.y - b.y, a.z - b.z, a.w - b.w);
}
__device__ __forceinline__ float4 f4add(float4 a, float4 b) {
  return make_float4(a.x + b.x, a.y + b.y, a.z + b.z, a.w + b.w);
}
__device__ __forceinline__ float4 f4scl(float4 a, float s) {
  return make_float4(a.x * s, a.y * s, a.z * s, a.w * s);
}
// value at col+1 per lane; for the global last column return cur.w so diff==0
__device__ __forceinline__ float4 shiftL1(float4 cur, float4 right, bool lastC) {
  return make_float4(cur.y, cur.z, cur.w, lastC ? cur.w : right.x);
}
// value at col-1 per lane; for global col 0 return cur.x so diff==0
__device__ __forceinline__ float4 shiftR1(float4 cur, float4 left, bool firstC) {
  return make_float4(firstC ? cur.x : left.w, cur.x, cur.y, cur.z);
}

// CDNA5 async data mover: copy global -> LDS without touching VGPRs.
// vdst = LDS byte offset, vaddr = 64-bit global address, saddr = off.
// Tracked by ASYNCcnt; consumer must s_wait_asynccnt then barrier.
__device__ __forceinline__ void async_cp_b128(const void* gsrc, void* lds_dst) {
  unsigned lds_off = (unsigned)(uintptr_t)lds_dst;  // low 32 bits = LDS offset
  asm volatile("global_load_async_to_lds_b128 %0, %1, off"
               :: "v"(lds_off), "v"(gsrc) : "memory");
}
__device__ __forceinline__ void async_cp_b32(const void* gsrc, void* lds_dst) {
  unsigned lds_off = (unsigned)(uintptr_t)lds_dst;
  asm volatile("global_load_async_to_lds_b32 %0, %1, off"
               :: "v"(lds_off), "v"(gsrc) : "memory");
}
__device__ __forceinline__ void async_wait0() {
  asm volatile("s_wait_asynccnt 0x0" ::: "memory");
}

__device__ __forceinline__ void grid_barrier(unsigned* cnt, unsigned* gen,
                                             unsigned nblocks) {
  __builtin_amdgcn_fence(__ATOMIC_RELEASE, "agent");
  __syncthreads();
  if (threadIdx.x == 0) {
    unsigned g = __hip_atomic_load(gen, __ATOMIC_RELAXED, __HIP_MEMORY_SCOPE_AGENT);
    unsigned prev = __hip_atomic_fetch_add(cnt, 1u, __ATOMIC_RELAXED,
                                           __HIP_MEMORY_SCOPE_AGENT);
    if (prev == nblocks - 1u) {
      __hip_atomic_store(cnt, 0u, __ATOMIC_RELAXED, __HIP_MEMORY_SCOPE_AGENT);
      __hip_atomic_fetch_add(gen, 1u, __ATOMIC_RELEASE, __HIP_MEMORY_SCOPE_AGENT);
    } else {
      while (__hip_atomic_load(gen, __ATOMIC_RELAXED,
                               __HIP_MEMORY_SCOPE_AGENT) == g) {
        __builtin_amdgcn_s_sleep(2);
      }
    }
  }
  __syncthreads();
  __builtin_amdgcn_fence(__ATOMIC_ACQUIRE, "agent");
}

// Copy/zero all state each call (harness poisons ws/out; graph replays us).
__global__ void tgv_init_kernel(const float* __restrict__ u_in,
                                const float* __restrict__ p_in,
                                float* __restrict__ ws,
                                unsigned* __restrict__ acc,
                                float* __restrict__ out) {
  const int idx = blockIdx.x * blockDim.x + threadIdx.x;
  const int stride = gridDim.x * blockDim.x;
  float* U   = ws;
  float* UB  = ws + (size_t)TGV_N2;
  float* P0  = ws + (size_t)2 * TGV_N2;
  float* P1  = ws + (size_t)3 * TGV_N2;
  float* PB0 = ws + (size_t)4 * TGV_N2;
  float* PB1 = ws + (size_t)5 * TGV_N2;
  float* VC0 = ws + (size_t)6 * TGV_N2;
  float* VC1 = ws + (size_t)7 * TGV_N2;
  float* W00 = ws + (size_t)8 * TGV_N2;
  float* W01 = ws + (size_t)9 * TGV_N2;
  float* W10 = ws + (size_t)10 * TGV_N2;
  float* W11 = ws + (size_t)11 * TGV_N2;
  for (int k = idx; k < TGV_N2; k += stride) {
    float u = u_in[k];
    U[k] = u; UB[k] = u; out[k] = u;   // out pre-filled: handles num_iters==0
    float p0 = p_in[2 * k + 0];
    float p1 = p_in[2 * k + 1];
    P0[k] = p0; P1[k] = p1; PB0[k] = p0; PB1[k] = p1;
    VC0[k] = 0.f; VC1[k] = 0.f;
    W00[k] = 0.f; W01[k] = 0.f; W10[k] = 0.f; W11[k] = 0.f;
  }
  if (idx < 8) acc[idx] = 0u;
}

__global__ __launch_bounds__(NTHREADS)
void tgv_pdhg_persistent(const float* __restrict__ u0_in,
                         const float* __restrict__ a1p,
                         const float* __restrict__ a0p,
                         const int* __restrict__ itp,
                         float* __restrict__ ws,
                         unsigned* __restrict__ acc,
                         float* __restrict__ out) {
  float* U   = ws;
  float* UB  = ws + (size_t)TGV_N2;
  float* P0  = ws + (size_t)2 * TGV_N2;
  float* P1  = ws + (size_t)3 * TGV_N2;
  float* PB0 = ws + (size_t)4 * TGV_N2;
  float* PB1 = ws + (size_t)5 * TGV_N2;
  float* VC0 = ws + (size_t)6 * TGV_N2;
  float* VC1 = ws + (size_t)7 * TGV_N2;
  float* W00 = ws + (size_t)8 * TGV_N2;
  float* W01 = ws + (size_t)9 * TGV_N2;
  float* W10 = ws + (size_t)10 * TGV_N2;
  float* W11 = ws + (size_t)11 * TGV_N2;
  unsigned* bar_cnt = acc + 4;
  unsigned* bar_gen = acc + 5;

  const float alpha1 = *a1p;
  const float alpha0 = *a0p;
  const int   T      = *itp;

  const int r0 = blockIdx.x * RPB;   // first row of this block's strip

  __shared__ __align__(16) float s_stage[STAGE_TOT];  // 36 KB: UB|P0|P1 rows
  __shared__ float s_mv[NTHREADS / 32];
  __shared__ float s_mw[NTHREADS / 32];
  __shared__ unsigned s_scraw[2];

  const float* sUB = s_stage;
  const float* sP0 = s_stage + STAGE_F;
  const float* sP1 = s_stage + 2 * STAGE_F;

#if TGV_USE_TDM
  // --- Tensor DMA descriptors (loop-invariant per block) -------------------
  // D# group0: count=1, lds_addr[63:32], global_addr[120:64], type=2[127:126]
  // D# group1: data_size=4B, tensor_dim0=768, tensor_dim1=rows-left (bottom
  //   block's 4th tile row is OOB -> zero-filled; dead behind lastR guard),
  //   tile_dim0=768, tile_dim1=4, tensor_dim0_stride=768. 2-D tile: remaining
  //   descriptor groups zero.
  const unsigned tdim1 = (unsigned)(TGV_N - r0);
  i32x8 g1v;
  g1v[0] = (int)(2u << 16);                                   // data_size=4B
  g1v[1] = (int)(((unsigned)TGV_N & 0xFFFFu) << 16);          // dim0 lo16
  g1v[2] = (int)(((unsigned)TGV_N >> 16) | ((tdim1 & 0xFFFFu) << 16));
  g1v[3] = (int)(((tdim1 >> 16) & 0xFFFFu) | ((unsigned)TGV_N << 16)); // tile0
  g1v[4] = (int)STAGE_ROWS;                                   // tile1=4,tile2=0
  g1v[5] = (int)TGV_N;                                        // stride0 lo32
  g1v[6] = 0;
  g1v[7] = 0;
  const i32x4 gzero4 = {0, 0, 0, 0};
  const i32x8 gzero8 = {0, 0, 0, 0, 0, 0, 0, 0};
  auto make_g0 = [](const float* gbase, unsigned lds_off) -> u32x4 {
    unsigned long long ga = (unsigned long long)(uintptr_t)gbase;
    u32x4 g;
    g.x = 1u;                                                 // count=1
    g.y = lds_off;
    g.z = (unsigned)ga;
    g.w = ((unsigned)(ga >> 32) & 0x01FFFFFFu) | (2u << 30);  // addr hi + type=2
    return g;
  };
  const u32x4 g0_ub = make_g0(UB + (size_t)r0 * TGV_N,
                              (unsigned)(uintptr_t)s_stage);
  const u32x4 g0_p0 = make_g0(P0 + (size_t)r0 * TGV_N,
                              (unsigned)(uintptr_t)(s_stage + STAGE_F));
  const u32x4 g0_p1 = make_g0(P1 + (size_t)r0 * TGV_N,
                              (unsigned)(uintptr_t)(s_stage + 2 * STAGE_F));
#endif

  float inv_svp = 1.0f;   // 1/scale_v of previous iteration (v_prev is UNscaled)
  float inv_swp = 1.0f;

  for (int t = 0; t < T; ++t) {
    // ------------- phase 1a: stage UB/P0/P1 strip (+halo row) into LDS -----
#if TGV_USE_TDM
    if (threadIdx.x < 32) {       // wave 0 issues the DMA, waits TENSORcnt
      __builtin_amdgcn_tensor_load_to_lds(g0_ub, g1v, gzero4, gzero4, gzero8, 0);
      __builtin_amdgcn_tensor_load_to_lds(g0_p0, g1v, gzero4, gzero4, gzero8, 0);
      __builtin_amdgcn_tensor_load_to_lds(g0_p1, g1v, gzero4, gzero4, gzero8, 0);
      __builtin_amdgcn_s_wait_tensorcnt(0);
    }
    __syncthreads();
#else
    for (int c = threadIdx.x; c < STAGE_CHUNKS; c += NTHREADS) {
      const int f   = c / (STAGE_F / 4);            // field 0..2
      const int rem = c - f * (STAGE_F / 4);
      const int lr  = rem / TGV_GJ;                 // staged row 0..3
      const int cg  = rem - lr * TGV_GJ;            // col group
      int gr = r0 + lr;
      gr = (gr < TGV_N) ? gr : (TGV_N - 1);         // clamped halo: dead data
      const float* gsrc = (f == 0 ? UB : (f == 1 ? P0 : P1)) +
                          (size_t)gr * TGV_N + (size_t)cg * 4;
      async_cp_b128(gsrc, s_stage + (size_t)c * 4);
    }
    async_wait0();
    __syncthreads();
#endif

    // ------------- phase 1b: dual candidates + global max norms ------------
    float mv = 0.f, mw = 0.f;
    for (int gl = threadIdx.x; gl < GPB; gl += NTHREADS) {
      const int li = gl / TGV_GJ;                   // local row 0..2
      const int jg = gl - li * TGV_GJ;
      const int i  = r0 + li;
      const bool lastR = (i == TGV_N - 1);
      const bool lastC = (jg == TGV_GJ - 1);

      // v_cand = v_prev/scale_prev + grad(u_bar) - p_bar     (SIGMA == 1)
      const float* rUB = sUB + li * TGV_N;
      float4 ub  = lds4(rUB + jg * 4);
      float4 ubn = lastC ? ub : lds4(rUB + jg * 4 + 4);
      float4 ubd = lastR ? ub : lds4(rUB + TGV_N + jg * 4);
      float4 g0  = f4sub(ubd, ub);                     // 0 on last row
      float4 g1  = f4sub(shiftL1(ub, ubn, lastC), ub); // 0 on last col
      float4 pb0 = ld4(PB0, i, jg);
      float4 pb1 = ld4(PB1, i, jg);
      float4 v0  = f4add(f4scl(ld4(VC0, i, jg), inv_svp), f4sub(g0, pb0));
      float4 v1  = f4add(f4scl(ld4(VC1, i, jg), inv_svp), f4sub(g1, pb1));
      st4(VC0, i, jg, v0);
      st4(VC1, i, jg, v1);
      {
        float n0 = fmaf(v0.x, v0.x, v1.x * v1.x);
        float n1 = fmaf(v0.y, v0.y, v1.y * v1.y);
        float n2 = fmaf(v0.z, v0.z, v1.z * v1.z);
        float n3 = fmaf(v0.w, v0.w, v1.w * v1.w);
        mv = fmaxf(mv, fmaxf(fmaxf(n0, n1), fmaxf(n2, n3)));
      }

      // w_cand = w_prev/scale_prev + grad(p)                 (SIGMA == 1)
      const float* rP0 = sP0 + li * TGV_N;
      const float* rP1 = sP1 + li * TGV_N;
      float4 p0  = lds4(rP0 + jg * 4);
      float4 p0n = lastC ? p0 : lds4(rP0 + jg * 4 + 4);
      float4 p0d = lastR ? p0 : lds4(rP0 + TGV_N + jg * 4);
      float4 gp00 = f4sub(p0d, p0);
      float4 gp01 = f4sub(shiftL1(p0, p0n, lastC), p0);
      float4 p1  = lds4(rP1 + jg * 4);
      float4 p1n = lastC ? p1 : lds4(rP1 + jg * 4 + 4);
      float4 p1d = lastR ? p1 : lds4(rP1 + TGV_N + jg * 4);
      float4 gp10 = f4sub(p1d, p1);
      float4 gp11 = f4sub(shiftL1(p1, p1n, lastC), p1);
      float4 w00 = f4add(f4scl(ld4(W00, i, jg), inv_swp), gp00);
      float4 w01 = f4add(f4scl(ld4(W01, i, jg), inv_swp), gp01);
      float4 w10 = f4add(f4scl(ld4(W10, i, jg), inv_swp), gp10);
      float4 w11 = f4add(f4scl(ld4(W11, i, jg), inv_swp), gp11);
      st4(W00, i, jg, w00);
      st4(W01, i, jg, w01);
      st4(W10, i, jg, w10);
      st4(W11, i, jg, w11);
      {
        float q0 = w00.x * w00.x + w01.x * w01.x + w10.x * w10.x + w11.x * w11.x;
        float q1 = w00.y * w00.y + w01.y * w01.y + w10.y * w10.y + w11.y * w11.y;
        float q2 = w00.z * w00.z + w01.z * w01.z + w10.z * w10.z + w11.z * w11.z;
        float q3 = w00.w * w00.w + w01.w * w01.w + w10.w * w10.w + w11.w * w11.w;
        mw = fmaxf(mw, fmaxf(fmaxf(q0, q1), fmaxf(q2, q3)));
      }
    }

    // block reduction (wave32 shuffle + LDS) then one atomic per field
    #pragma unroll
    for (int off = 16; off > 0; off >>= 1) {
      mv = fmaxf(mv, __shfl_down(mv, off, 32));
      mw = fmaxf(mw, __shfl_down(mw, off, 32));
    }
    const int lane = threadIdx.x & 31;
    const int wv   = threadIdx.x >> 5;
    if (lane == 0) { s_mv[wv] = mv; s_mw[wv] = mw; }
    __syncthreads();
    if (threadIdx.x == 0) {
      float a = s_mv[0], b = s_mw[0];
      #pragma unroll
      for (int k = 1; k < NTHREADS / 32; ++k) {
        a = fmaxf(a, s_mv[k]);
        b = fmaxf(b, s_mw[k]);
      }
      // norms are >= 0, so the uint bit pattern is order-preserving
      atomicMax(acc + (t & 1), __float_as_uint(a));
      atomicMax(acc + 2 + (t & 1), __float_as_uint(b));
    }

    grid_barrier(bar_cnt, bar_gen, (unsigned)gridDim.x);

    // ------------- phase 2: projection scales + primal updates -------------
    // async-broadcast the two global max words straight into LDS
    if (threadIdx.x < 2) {
      async_cp_b32(acc + (t & 1) + 2 * threadIdx.x, &s_scraw[threadIdx.x]);
    }
    async_wait0();
    __syncthreads();
    const float gmv = __uint_as_float(s_scraw[0]);
    const float gmw = __uint_as_float(s_scraw[1]);
    const float inv_sv = 1.0f / fmaxf(1.f, sqrtf(gmv) / alpha1);
    const float inv_sw = 1.0f / fmaxf(1.f, sqrtf(gmw) / alpha0);

    for (int gl = threadIdx.x; gl < GPB; gl += NTHREADS) {
      const int li = gl / TGV_GJ;
      const int jg = gl - li * TGV_GJ;
      const int i  = r0 + li;
      const bool firstR = (i == 0);
      const bool firstC = (jg == 0);

      const int gn = gl + NTHREADS;
      if (gn < GPB) {
        const int li2 = gn / TGV_GJ, j2 = gn - li2 * TGV_GJ;
        __builtin_prefetch(reinterpret_cast<const float4*>(
                               VC0 + (size_t)(r0 + li2) * TGV_N) + j2, 0, 0);
      }

      // u_n = (u + div(v_n) + u0) / 2          (TAU == 1)
      float4 v0  = ld4(VC0, i, jg);
      float4 v0u = firstR ? v0 : ld4(VC0, i - 1, jg);
      float4 v1  = ld4(VC1, i, jg);
      float4 v1l = firstC ? v1 : ld4(VC1, i, jg - 1);
      float4 d0  = f4sub(v0, v0u);                       // 0 on first row
      float4 d1  = f4sub(v1, shiftR1(v1, v1l, firstC));  // 0 on first col
      float4 u   = ld4(U, i, jg);
      float4 uo  = ld4(u0_in, i, jg);
      float4 un;
      un.x = (u.x + (d0.x + d1.x) * inv_sv + uo.x) * 0.5f;
      un.y = (u.y + (d0.y + d1.y) * inv_sv + uo.y) * 0.5f;
      un.z = (u.z + (d0.z + d1.z) * inv_sv + uo.z) * 0.5f;
      un.w = (u.w + (d0.w + d1.w) * inv_sv + uo.w) * 0.5f;
      st4(U, i, jg, un);
      st4(UB, i, jg, make_float4(2.f * un.x - u.x, 2.f * un.y - u.y,
                                 2.f * un.z - u.z, 2.f * un.w - u.w));
      if (t == T - 1) st4(out, i, jg, un);

      // p_n = p + v_n + div(w_n)               (TAU == 1)
      float4 w00  = ld4(W00, i, jg);
      float4 w00u = firstR ? w00 : ld4(W00, i - 1, jg);
      float4 w01  = ld4(W01, i, jg);
      float4 w01l = firstC ? w01 : ld4(W01, i, jg - 1);
      float4 dw0  = f4add(f4sub(w00, w00u),
                          f4sub(w01, shiftR1(w01, w01l, firstC)));
      float4 w10  = ld4(W10, i, jg);
      float4 w10u = firstR ? w10 : ld4(W10, i - 1, jg);
      float4 w11  = ld4(W11, i, jg);
      float4 w11l = firstC ? w11 : ld4(W11, i, jg - 1);
      float4 dw1  = f4add(f4sub(w10, w10u),
                          f4sub(w11, shiftR1(w11, w11l, firstC)));

      float4 p0 = ld4(P0, i, jg);
      float4 p0n;
      p0n.x = p0.x + v0.x * inv_sv + dw0.x * inv_sw;
      p0n.y = p0.y + v0.y * inv_sv + dw0.y * inv_sw;
      p0n.z = p0.z + v0.z * inv_sv + dw0.z * inv_sw;
      p0n.w = p0.w + v0.w * inv_sv + dw0.w * inv_sw;
      st4(P0, i, jg, p0n);
      st4(PB0, i, jg, make_float4(2.f * p0n.x - p0.x, 2.f * p0n.y - p0.y,
                                  2.f * p0n.z - p0.z, 2.f * p0n.w - p0.w));

      float4 p1 = ld4(P1, i, jg);
      float4 p1n;
      p1n.x = p1.x + v1.x * inv_sv + dw1.x * inv_sw;
      p1n.y = p1.y + v1.y * inv_sv + dw1.y * inv_sw;
      p1n.z = p1.z + v1.z * inv_sv + dw1.z * inv_sw;
      p1n.w = p1.w + v1.w * inv_sv + dw1.w * inv_sw;
      st4(P1, i, jg, p1n);
      st4(PB1, i, jg, make_float4(2.f * p1n.x - p1.x, 2.f * p1n.y - p1.y,
                                  2.f * p1n.z - p1.z, 2.f * p1n.w - p1.w));
    }

    // reset the *other* ping-pong slots for the next iteration
    if (blockIdx.x == 0 && threadIdx.x == 0) {
      __hip_atomic_store(acc + ((t + 1) & 1), 0u, __ATOMIC_RELAXED,
                         __HIP_MEMORY_SCOPE_AGENT);
      __hip_atomic_store(acc + 2 + ((t + 1) & 1), 0u, __ATOMIC_RELAXED,
                         __HIP_MEMORY_SCOPE_AGENT);
    }
    inv_svp = inv_sv;
    inv_swp = inv_sw;

    grid_barrier(bar_cnt, bar_gen, (unsigned)gridDim.x);
  }
}

extern "C" void kernel_launch(void* const* d_in, const int* in_sizes, int n_in,
                              void* d_out, int out_size, void* d_ws, size_t ws_size,
                              hipStream_t stream) {
  (void)in_sizes; (void)n_in; (void)out_size; (void)ws_size;
  const float* u  = (const float*)d_in[0];
  const float* p  = (const float*)d_in[1];
  const float* a1 = (const float*)d_in[2];
  const float* a0 = (const float*)d_in[3];
  const int*   ni = (const int*)d_in[4];
  float* out = (float*)d_out;
  float* ws  = (float*)d_ws;
  unsigned* acc = (unsigned*)(ws + (size_t)12 * TGV_N2);

  tgv_init_kernel<<<576, NTHREADS, 0, stream>>>(u, p, ws, acc, out);
  tgv_pdhg_persistent<<<NBLOCKS, NTHREADS, 0, stream>>>(u, a1, a0, ni, ws, acc, out);
}